// PCFormer_76038101009026
// MI455X (gfx1250) — compile-verified
//
#include <hip/hip_runtime.h>
#include <hip/hip_bf16.h>

// ---------------- problem constants (match reference) ----------------
#define NN   100000
#define EE   3200000
#define D_IN 256
#define D_H  128
#define HH   2        // heads; q/k/xs are [N, 256] with head h in cols [h*128, h*128+128)
#define D_O  40
#define BETA 0.5f
#define EPS  1e-5f

typedef __attribute__((ext_vector_type(2))) float v2f;
typedef __attribute__((ext_vector_type(8))) float v8f;

// =====================================================================
// Generic fp32 WMMA GEMM: C = alpha * (A@B + bias), A:[M,K] lda, B:[K,N] ldb.
// Requirement: K % 16 == 0 (true for all GEMMs in this model: K in {128,256}).
// Block = 256 threads = 8 waves; block tile 128(M) x 64(N); K-chunk 16.
// Wave grid 4x2; each wave computes 32x32 via 2x2 v_wmma_f32_16x16x4_f32.
// Staging loads are branchless (index clamp + cndmask zero-fill).
// =====================================================================
__global__ __launch_bounds__(256)
void gemm_wmma_f32(const float* __restrict__ A, const float* __restrict__ B,
                   const float* __restrict__ bias, float* __restrict__ C,
                   int M, int N, int K, int lda, int ldb, int ldc, float alpha)
{
    __shared__ float sA[128 * 18];   // 128 rows x 16 cols, stride 18 (pad)
    __shared__ float sB[16 * 65];    // 16 rows x 64 cols, stride 65 (pad)
    const int tid  = threadIdx.x;
    const int wave = tid >> 5;
    const int lane = tid & 31;
    const int m0 = blockIdx.x * 128;
    const int n0 = blockIdx.y * 64;
    const int wm = (wave >> 1) * 32;   // wave row offset in block tile
    const int wc = (wave & 1) * 32;    // wave col offset in block tile
    const int mf = lane & 15;
    const int kb = (lane >> 4) * 2;

    // per-thread staging coordinates (8x A elems, 4x B elems per chunk)
    const int ar = tid >> 4;          // A rows handled: ar, ar+16, ..., ar+112
    const int ac = tid & 15;          // A col within chunk
    const int br = tid >> 6;          // B rows handled: br, br+4, br+8, br+12
    const int bc = tid & 63;          // B col within tile

    v8f acc[2][2] = {};

    for (int k0 = 0; k0 < K; k0 += 16) {
        // prefetch next A chunk (global_prefetch_b8); harmless at K edge (clamped row)
        {
            int pr = min(m0 + ar, M - 1);
            int pk = k0 + 16 < K ? k0 + 16 : k0;
            __builtin_prefetch(&A[(long)pr * lda + pk + ac], 0, 0);
        }
        // stage A 128x16 (branchless: clamp row, zero-fill OOB)
        #pragma unroll
        for (int i = 0; i < 8; ++i) {
            int r = ar + i * 16;
            int gr = m0 + r;
            float v = A[(long)min(gr, M - 1) * lda + k0 + ac];
            sA[r * 18 + ac] = (gr < M) ? v : 0.f;
        }
        // stage B 16x64 (branchless: clamp col, zero-fill OOB; K always in range)
        #pragma unroll
        for (int i = 0; i < 4; ++i) {
            int r = br + i * 4;
            int gc = n0 + bc;
            float v = B[(long)(k0 + r) * ldb + min(gc, N - 1)];
            sB[r * 65 + bc] = (gc < N) ? v : 0.f;
        }
        __syncthreads();
        #pragma unroll
        for (int kk = 0; kk < 16; kk += 4) {
            v2f af[2], bf[2];
            #pragma unroll
            for (int i = 0; i < 2; ++i) {
                int r = wm + i * 16 + mf;
                af[i].x = sA[r * 18 + kk + kb];
                af[i].y = sA[r * 18 + kk + kb + 1];
            }
            #pragma unroll
            for (int j = 0; j < 2; ++j) {
                int c = wc + j * 16 + mf;
                bf[j].x = sB[(kk + kb) * 65 + c];
                bf[j].y = sB[(kk + kb + 1) * 65 + c];
            }
            #pragma unroll
            for (int i = 0; i < 2; ++i)
                #pragma unroll
                for (int j = 0; j < 2; ++j)
                    acc[i][j] = __builtin_amdgcn_wmma_f32_16x16x4_f32(
                        false, af[i], false, bf[j], (short)0, acc[i][j], false, false);
        }
        __syncthreads();
    }

    const int rg = (lane >> 4) * 8;   // C/D layout: lanes 16-31 hold rows 8..15
    const int cf = lane & 15;
    #pragma unroll
    for (int i = 0; i < 2; ++i)
        #pragma unroll
        for (int j = 0; j < 2; ++j) {
            int cg = n0 + wc + j * 16 + cf;
            if (cg >= N) continue;
            float bv = bias ? bias[cg] : 0.f;
            #pragma unroll
            for (int t = 0; t < 8; ++t) {
                int rgl = m0 + wm + i * 16 + rg + t;
                if (rgl < M) C[(long)rgl * ldc + cg] = alpha * (acc[i][j][t] + bv);
            }
        }
}

// =====================================================================
// kvs[h][m,d] = sum_n k[n, h*128+m] * xs[n, h*128+d]   (A^T B, split-K atomic)
// Block = 256 thr = 8 waves; wave w owns output rows [w*16, w*16+16) x 128 cols.
// =====================================================================
__global__ __launch_bounds__(256)
void atb_kvs(const float* __restrict__ k, const float* __restrict__ xs,
             float* __restrict__ kvs, int n_rows, int rowsPerBlock)
{
    __shared__ float sK[16 * 132];
    __shared__ float sX[16 * 132];
    const int h = blockIdx.y;
    const int tid = threadIdx.x, wave = tid >> 5, lane = tid & 31;
    const int mf = lane & 15, kb = (lane >> 4) * 2;
    const int rowBeg = blockIdx.x * rowsPerBlock;
    const int rowEnd = min(rowBeg + rowsPerBlock, n_rows);
    const int lr = tid >> 7;          // local rows handled: lr, lr+2, ..., lr+14
    const int lc = tid & 127;

    v8f acc[8] = {};

    for (int r0 = rowBeg; r0 < rowEnd; r0 += 16) {
        #pragma unroll
        for (int i = 0; i < 8; ++i) {
            int rr = lr + i * 2;
            int gr = r0 + rr;
            int grc = min(gr, rowEnd - 1);
            float vk = k[(long)grc * 256 + h * 128 + lc];
            float vx = xs[(long)grc * 256 + h * 128 + lc];
            bool ok = gr < rowEnd;
            sK[rr * 132 + lc] = ok ? vk : 0.f;
            sX[rr * 132 + lc] = ok ? vx : 0.f;
        }
        __syncthreads();
        #pragma unroll
        for (int kk = 0; kk < 16; kk += 4) {
            v2f a;
            a.x = sK[(kk + kb) * 132 + wave * 16 + mf];       // A[m,nn] = k_h[nn,m]
            a.y = sK[(kk + kb + 1) * 132 + wave * 16 + mf];
            #pragma unroll
            for (int j = 0; j < 8; ++j) {
                v2f b;
                b.x = sX[(kk + kb) * 132 + j * 16 + mf];
                b.y = sX[(kk + kb + 1) * 132 + j * 16 + mf];
                acc[j] = __builtin_amdgcn_wmma_f32_16x16x4_f32(
                    false, a, false, b, (short)0, acc[j], false, false);
            }
        }
        __syncthreads();
    }
    const int rg = (lane >> 4) * 8, cf = lane & 15;
    #pragma unroll
    for (int j = 0; j < 8; ++j)
        #pragma unroll
        for (int t = 0; t < 8; ++t)
            atomicAdd(&kvs[(long)h * 16384 + (wave * 16 + rg + t) * 128 + j * 16 + cf],
                      acc[j][t]);
}

// ---------------- graph weight kernels ----------------
__global__ void degree_k(const int* __restrict__ ei, float* __restrict__ deg, int E)
{
    int e = blockIdx.x * 256 + threadIdx.x;
    if (e < E) atomicAdd(&deg[ei[E + e]], 1.0f);   // count over col
}

__global__ void edgew_k(const int* __restrict__ ei, const float* __restrict__ deg,
                        float* __restrict__ wn, int E)
{
    int e = blockIdx.x * 256 + threadIdx.x;
    if (e < E) {
        float dr = deg[ei[e]], dc = deg[ei[E + e]];
        float a = dr > 0.f ? rsqrtf(dr) : 0.f;
        float b = dc > 0.f ? rsqrtf(dc) : 0.f;
        wn[e] = a * b;
    }
}

// SpMM: out[col] += wn[e] * xs[row], 256-wide features; one thread = one edge x 4 cols.
__global__ void spmm_edge(const int* __restrict__ ei, const float* __restrict__ wn,
                          const float* __restrict__ xs, float* __restrict__ out, int E)
{
    long idx = (long)blockIdx.x * 256 + threadIdx.x;
    if (idx >= (long)E * 64) return;
    int e = (int)(idx >> 6);
    int j = (int)(idx & 63) << 2;
    int r = ei[e], c = ei[E + e];
    float w = wn[e];
    float4 v = *(const float4*)(xs + (long)r * 256 + j);
    float* o = out + (long)c * 256 + j;
    atomicAdd(o + 0, w * v.x);
    atomicAdd(o + 1, w * v.y);
    atomicAdd(o + 2, w * v.z);
    atomicAdd(o + 3, w * v.w);
}

// ---------------- BN (train-stats) ----------------
__global__ void bn_stats(const float* __restrict__ X, float* __restrict__ sum,
                         float* __restrict__ sq, int M)
{
    int c = threadIdx.x & 127;
    int rh = threadIdx.x >> 7;
    float s = 0.f, qq = 0.f;
    for (int r = blockIdx.x * 2 + rh; r < M; r += gridDim.x * 2) {
        float v = X[(long)r * 128 + c];
        s += v; qq += v * v;
    }
    __shared__ float ls[256], lq[256];
    ls[threadIdx.x] = s; lq[threadIdx.x] = qq;
    __syncthreads();
    if (threadIdx.x < 128) {
        atomicAdd(&sum[c], ls[threadIdx.x] + ls[threadIdx.x + 128]);
        atomicAdd(&sq[c],  lq[threadIdx.x] + lq[threadIdx.x + 128]);
    }
}

__global__ void bn_apply_relu(const float* __restrict__ Xin, float* __restrict__ Xout,
                              const float* __restrict__ sum, const float* __restrict__ sq,
                              const float* __restrict__ g, const float* __restrict__ b, int M)
{
    long i = (long)blockIdx.x * 256 + threadIdx.x;
    if (i >= (long)M * 128) return;
    int c = (int)(i & 127);
    float inv = 1.0f / (float)M;
    float m = sum[c] * inv;
    float v = sq[c] * inv - m * m;
    float y = g[c] * (Xin[i] - m) * rsqrtf(v + EPS) + b[c];
    Xout[i] = fmaxf(y, 0.f);
}

// ---------------- q/k row L2-normalize: one wave per 128-float segment ----------------
__global__ void l2norm128(float* __restrict__ X, long rows)
{
    long w = ((long)blockIdx.x * blockDim.x + threadIdx.x) >> 5;
    int lane = threadIdx.x & 31;
    if (w >= rows) return;
    float* p = X + w * 128;
    float4 v = *(float4*)(p + lane * 4);
    float s = v.x * v.x + v.y * v.y + v.z * v.z + v.w * v.w;
    for (int off = 16; off; off >>= 1) s += __shfl_xor(s, off, 32);
    float inv = rsqrtf(s);
    v.x *= inv; v.y *= inv; v.z *= inv; v.w *= inv;
    *(float4*)(p + lane * 4) = v;
}

// ---------------- misc elementwise ----------------
__global__ void col_sum256(const float* __restrict__ X, float* __restrict__ out, int M)
{
    int c = threadIdx.x;       // blockDim == 256
    float s = 0.f;
    for (int r = blockIdx.x; r < M; r += gridDim.x) s += X[(long)r * 256 + c];
    atomicAdd(&out[c], s);
}

__global__ void init_xs(const float* __restrict__ h0, float* __restrict__ xs, int M)
{
    long i = (long)blockIdx.x * 256 + threadIdx.x;
    if (i >= (long)M * 128) return;
    long n = i >> 7; int d = (int)(i & 127);
    float v = h0[i];
    xs[n * 256 + d] = v;
    xs[n * 256 + 128 + d] = v;
}

__global__ void axpy_k(float* __restrict__ y, const float* __restrict__ x, float a, long n)
{
    long i = (long)blockIdx.x * 256 + threadIdx.x;
    if (i < n) y[i] += a * x[i];
}

__global__ void add_k(float* __restrict__ y, const float* __restrict__ x, long n)
{
    long i = (long)blockIdx.x * 256 + threadIdx.x;
    if (i < n) y[i] += x[i];
}

// xs[n,h,d] += (1-BETA) * (numq[n,h,d] + xs_sum[h,d]) / (q[n,h,:].ks_sum[h,:] + N)
__global__ __launch_bounds__(256)
void attn_update(float* __restrict__ xs, const float* __restrict__ numq,
                 const float* __restrict__ q, const float* __restrict__ kss,
                 const float* __restrict__ xss, int n_nodes)
{
    __shared__ float red[256];
    int t = threadIdx.x;
    int d = t & 127;
    int hbase = t & 128;   // 0 or 128
    for (int n = blockIdx.x; n < n_nodes; n += gridDim.x) {
        float qv = q[(long)n * 256 + t];
        red[t] = qv * kss[t];
        __syncthreads();
        for (int off = 64; off; off >>= 1) {
            if (d < off) red[t] += red[t + off];
            __syncthreads();
        }
        float den = red[hbase] + (float)NN;
        __syncthreads();
        float attn = (numq[(long)n * 256 + t] + xss[t]) / den;
        xs[(long)n * 256 + t] += (1.0f - BETA) * attn;
    }
}

// =====================================================================
extern "C" void kernel_launch(void* const* d_in, const int* in_sizes, int n_in,
                              void* d_out, int out_size, void* d_ws, size_t ws_size,
                              hipStream_t stream)
{
    const float* x     = (const float*)d_in[0];
    const float* W_in  = (const float*)d_in[1];
    const float* b_in  = (const float*)d_in[2];
    const float* g0    = (const float*)d_in[3];
    const float* be0   = (const float*)d_in[4];
    const float* Wq    = (const float*)d_in[5];
    const float* bq    = (const float*)d_in[6];
    const float* Wk    = (const float*)d_in[7];
    const float* bk    = (const float*)d_in[8];
    const float* Wo    = (const float*)d_in[9];
    const float* bo    = (const float*)d_in[10];
    const float* g1    = (const float*)d_in[11];
    const float* be1   = (const float*)d_in[12];
    const float* W_out = (const float*)d_in[13];
    const float* b_out = (const float*)d_in[14];
    const int*   ei    = (const int*)d_in[15];
    float* out = (float*)d_out;

    // workspace layout (floats); total ~131.4M floats (~526 MB)
    float* ws = (float*)d_ws;
    size_t off = 0;
    float* deg   = ws + off; off += 102400;
    float* wn    = ws + off; off += (size_t)EE;
    float* h0    = ws + off; off += (size_t)NN * 128;
    float* hb    = ws + off; off += (size_t)NN * 128;   // pre-BN / h1 buffer
    float* q     = ws + off; off += (size_t)NN * 256;
    float* k     = ws + off; off += (size_t)NN * 256;
    float* xs    = ws + off; off += (size_t)NN * 256;
    float* tmp   = ws + off; off += (size_t)NN * 256;   // gcn output, then q@kvs output
    float* kvs   = ws + off; off += 2 * 128 * 128;
    float* kssum = ws + off; off += 256;
    float* xssum = ws + off; off += 256;
    float* bns   = ws + off; off += 128;
    float* bnq   = ws + off; off += 128;

    const int gM = (NN + 127) / 128;                 // 782 M-tiles
    const dim3 blk256(256);
    const int eBlocks = (EE + 255) / 256;

    // ---- edge weights: deg over col, wn = rsqrt(deg[col])*rsqrt(deg[row])
    hipMemsetAsync(deg, 0, (size_t)NN * 4, stream);
    degree_k<<<eBlocks, blk256, 0, stream>>>(ei, deg, EE);
    edgew_k<<<eBlocks, blk256, 0, stream>>>(ei, deg, wn, EE);

    // ---- fc0: hb = x@W_in + b_in ; h0 = relu(bn(hb))
    gemm_wmma_f32<<<dim3(gM, 1), blk256, 0, stream>>>(x, W_in, b_in, hb,
        NN, D_H, D_IN, D_IN, D_H, D_H, 1.0f);
    hipMemsetAsync(bns, 0, 128 * 4, stream);
    hipMemsetAsync(bnq, 0, 128 * 4, stream);
    bn_stats<<<256, blk256, 0, stream>>>(hb, bns, bnq, NN);
    bn_apply_relu<<<(NN * 128 + 255) / 256, blk256, 0, stream>>>(hb, h0, bns, bnq, g0, be0, NN);

    // ---- q,k projections + per-head L2 normalize
    gemm_wmma_f32<<<dim3(gM, 4), blk256, 0, stream>>>(h0, Wq, bq, q,
        NN, 256, D_H, D_H, 256, 256, 1.0f);
    gemm_wmma_f32<<<dim3(gM, 4), blk256, 0, stream>>>(h0, Wk, bk, k,
        NN, 256, D_H, D_H, 256, 256, 1.0f);
    {
        long rows = (long)NN * HH;                  // 200k segments of 128
        int blocks = (int)((rows * 32 + 255) / 256);
        l2norm128<<<blocks, blk256, 0, stream>>>(q, rows);
        l2norm128<<<blocks, blk256, 0, stream>>>(k, rows);
    }

    // ---- ks_sum (fixed across iterations)
    hipMemsetAsync(kssum, 0, 256 * 4, stream);
    col_sum256<<<256, blk256, 0, stream>>>(k, kssum, NN);

    // ---- xs = repeat(h0, heads)
    init_xs<<<(NN * 128 + 255) / 256, blk256, 0, stream>>>(h0, xs, NN);

    // ---- K_ORDER propagation iterations
    const int kvsBlocks = 128;
    const int rowsPerBlock = (NN + kvsBlocks - 1) / kvsBlocks;
    for (int it = 0; it < 3; ++it) {
        hipMemsetAsync(tmp, 0, (size_t)NN * 256 * 4, stream);
        hipMemsetAsync(kvs, 0, 2 * 128 * 128 * 4, stream);
        hipMemsetAsync(xssum, 0, 256 * 4, stream);

        // gcn = prop(xs) into tmp
        {
            long work = (long)EE * 64;
            int blocks = (int)((work + 255) / 256);
            spmm_edge<<<blocks, blk256, 0, stream>>>(ei, wn, xs, tmp, EE);
        }
        // kvs and xs_sum from OLD xs
        atb_kvs<<<dim3(kvsBlocks, HH), blk256, 0, stream>>>(k, xs, kvs, NN, rowsPerBlock);
        col_sum256<<<256, blk256, 0, stream>>>(xs, xssum, NN);

        // xs += BETA * gcn   (old xs no longer needed elementwise)
        axpy_k<<<(NN * 256 + 255) / 256, blk256, 0, stream>>>(xs, tmp, BETA, (long)NN * 256);

        // tmp = q @ kvs per head (overwrites tmp)
        for (int h = 0; h < HH; ++h)
            gemm_wmma_f32<<<dim3(gM, 2), blk256, 0, stream>>>(
                q + h * 128, kvs + (size_t)h * 16384, nullptr, tmp + h * 128,
                NN, D_H, D_H, 256, D_H, 256, 1.0f);

        // xs += (1-BETA) * (tmp + xs_sum) / den
        attn_update<<<NN, blk256, 0, stream>>>(xs, tmp, q, kssum, xssum, NN);
    }

    // ---- hb = (xs@Wo + bo)/H ; hb += h0 ; hb = relu(bn(hb))
    gemm_wmma_f32<<<dim3(gM, 2), blk256, 0, stream>>>(xs, Wo, bo, hb,
        NN, D_H, 256, 256, D_H, D_H, 1.0f / (float)HH);
    add_k<<<(NN * 128 + 255) / 256, blk256, 0, stream>>>(hb, h0, (long)NN * 128);
    hipMemsetAsync(bns, 0, 128 * 4, stream);
    hipMemsetAsync(bnq, 0, 128 * 4, stream);
    bn_stats<<<256, blk256, 0, stream>>>(hb, bns, bnq, NN);
    bn_apply_relu<<<(NN * 128 + 255) / 256, blk256, 0, stream>>>(hb, hb, bns, bnq, g1, be1, NN);

    // ---- out = hb @ W_out + b_out   [N, 40]
    gemm_wmma_f32<<<dim3(gM, 1), blk256, 0, stream>>>(hb, W_out, b_out, out,
        NN, D_O, D_H, D_H, D_O, D_O, 1.0f);

    (void)in_sizes; (void)n_in; (void)out_size; (void)ws_size;
}